// WeightOptimizer_60679297958028
// MI455X (gfx1250) — compile-verified
//
#include <hip/hip_runtime.h>

// CDNA5 / gfx1250: fp32 WMMA (16x16x4), wave32.
typedef __attribute__((ext_vector_type(2))) float v2f;
typedef __attribute__((ext_vector_type(8))) float v8f;

#define T_DIM   8192
#define N_DIM   64
#define BT      8              // band tiles of 16 -> history depth 127 (g(d) underflows past ~127 in fp32)
#define WPB     4              // waves per block

#define E_M07   0.49658530379140951f   // exp(-0.7)
#define E_M28   0.060810062625217984f  // exp(-2.8) : per-K-step (K=4) decay of e = exp(0.7*(d-11))

// One 16x16x4 K-step: A = causal decay kernel g(k-i) (Toeplitz), B = w[i]*x[i][c].
// e0/e1 carry exp(0.7*(d-11)) via recurrence (updated by caller).
template <bool CLAMP, bool MASK>
__device__ __forceinline__ void kstep(
    const float* __restrict__ x, const float* __restrict__ w,
    int i0, int d0, float e0, float e1, float onesMask, int l16,
    v8f& acc0, v8f& acc1, v8f& acc2, v8f& acc3, v8f& accD)
{
    const int i1 = i0 + 1;

    // g(d) = 1/(1+exp(0.7*(d-11))) ; fast v_rcp_f32 (arg >= 1, ~1ulp is plenty)
    const float g0 = __builtin_amdgcn_rcpf(1.0f + e0);
    const float g1 = __builtin_amdgcn_rcpf(1.0f + e1);
    v2f A;
    A[0] = MASK ? ((d0 >= 1) ? g0 : 0.0f) : g0;   // d1 = d0-1
    A[1] = MASK ? ((d0 >= 2) ? g1 : 0.0f) : g1;

    int i0c = i0, i1c = i1;
    float w0, w1;
    if (CLAMP) {
        // branch-free: unconditional in-bounds load, arithmetic 0/1 mask
        i0c = (i0 > 0) ? i0 : 0;
        i1c = (i1 > 0) ? i1 : 0;
        const float m0 = (i0 >= 0) ? 1.0f : 0.0f;
        const float m1 = (i1 >= 0) ? 1.0f : 0.0f;
        w0 = w[i0c] * m0;
        w1 = w[i1c] * m1;
    } else {
        w0 = w[i0];
        w1 = w[i1];
    }

    const float* r0 = x + (size_t)i0c * N_DIM + l16;
    const float* r1 = x + (size_t)i1c * N_DIM + l16;

    v2f B0; B0[0] = w0 * r0[ 0]; B0[1] = w1 * r1[ 0];
    v2f B1; B1[0] = w0 * r0[16]; B1[1] = w1 * r1[16];
    v2f B2; B2[0] = w0 * r0[32]; B2[1] = w1 * r1[32];
    v2f B3; B3[0] = w0 * r0[48]; B3[1] = w1 * r1[48];
    v2f Bd; Bd[0] = onesMask * w0; Bd[1] = onesMask * w1;

    acc0 = __builtin_amdgcn_wmma_f32_16x16x4_f32(false, A, false, B0, (short)0, acc0, false, false);
    acc1 = __builtin_amdgcn_wmma_f32_16x16x4_f32(false, A, false, B1, (short)0, acc1, false, false);
    acc2 = __builtin_amdgcn_wmma_f32_16x16x4_f32(false, A, false, B2, (short)0, acc2, false, false);
    acc3 = __builtin_amdgcn_wmma_f32_16x16x4_f32(false, A, false, B3, (short)0, acc3, false, false);
    accD = __builtin_amdgcn_wmma_f32_16x16x4_f32(false, A, false, Bd, (short)0, accD, false, false);
}

__global__ __launch_bounds__(WPB * 32) void band_wmma_kernel(
    const float* __restrict__ x,
    const float* __restrict__ w,
    float* __restrict__ partial)
{
    const int lane    = threadIdx.x & 31;
    const int wave    = threadIdx.x >> 5;
    const int gw      = blockIdx.x * WPB + wave;   // global wave id = 16-row output tile
    const int rowBase = gw * 16;

    const int l16 = lane & 15;      // M (A) / N (B,D) index
    const int h   = lane >> 4;      // half-wave select (K pairing)
    const float onesMask = (l16 == 0) ? 1.0f : 0.0f;

    v8f acc0 = {}; v8f acc1 = {}; v8f acc2 = {}; v8f acc3 = {};
    v8f accD = {};

    // d at (t=0, ks=0) for this lane's K=2h element: d = l16 + 112 - 2h.
    // One v_exp per wave; thereafter e *= exp(-2.8) per K-step.
    float e0 = __expf(0.7f * ((float)(l16 + 112 - 2 * h) - 11.0f));
    float e1 = e0 * E_M07;
    const int iBand = rowBase - 16 * (BT - 1);

    if (rowBase >= 16 * (BT - 1)) {
        // steady state: band fully in-range, strictly below diagonal -> no masks at all
        #pragma unroll 1
        for (int t = 0; t < BT - 1; ++t) {
            #pragma unroll
            for (int ks = 0; ks < 4; ++ks) {
                const int i0 = iBand + 16 * t + 4 * ks + 2 * h;
                kstep<false, false>(x, w, i0, 0, e0, e1, onesMask, l16,
                                    acc0, acc1, acc2, acc3, accD);
                e0 *= E_M28; e1 *= E_M28;
            }
        }
    } else {
        // first 7 waves: band starts before row 0 -> clamp index + arithmetic weight mask
        #pragma unroll 1
        for (int t = 0; t < BT - 1; ++t) {
            #pragma unroll
            for (int ks = 0; ks < 4; ++ks) {
                const int i0 = iBand + 16 * t + 4 * ks + 2 * h;
                kstep<true, false>(x, w, i0, 0, e0, e1, onesMask, l16,
                                   acc0, acc1, acc2, acc3, accD);
                e0 *= E_M28; e1 *= E_M28;
            }
        }
    }

    // diagonal tile: i in [rowBase, rowBase+15] -> causal mask d>=1 (indices always in range)
    #pragma unroll
    for (int ks = 0; ks < 4; ++ks) {
        const int i0 = rowBase + 4 * ks + 2 * h;
        const int d0 = l16 - 4 * ks - 2 * h;
        kstep<false, true>(x, w, i0, d0, e0, e1, onesMask, l16,
                           acc0, acc1, acc2, acc3, accD);
        e0 *= E_M28; e1 *= E_M28;
    }

    // D layout: lane holds column l16, rows r + 8h (r = VGPR index).
    // den for row (r+8h) lives in lane 16h, element r -> broadcast via shfl.
    float sum = 0.0f;
    #pragma unroll
    for (int r = 0; r < 8; ++r) {
        const int   kr  = rowBase + r + 8 * h;
        const float den = __shfl(accD[r], 16 * h, 32);
        if (kr >= 1) {   // k = 0 has empty history; reference drops it
            const float  rden   = 1.0f / den;
            const size_t rowOff = (size_t)kr * N_DIM + l16;
            const float g0 = acc0[r] * rden - x[rowOff +  0];
            const float g1 = acc1[r] * rden - x[rowOff + 16];
            const float g2 = acc2[r] * rden - x[rowOff + 32];
            const float g3 = acc3[r] * rden - x[rowOff + 48];
            sum += g0 * g0 + g1 * g1 + g2 * g2 + g3 * g3;
        }
    }

    // deterministic wave32 reduction, one partial per wave
    #pragma unroll
    for (int off = 16; off >= 1; off >>= 1)
        sum += __shfl_xor(sum, off, 32);
    if (lane == 0) partial[gw] = sum;
}

// Fixed-topology tree reduction of the 512 per-wave partials -> scalar MSE.
__global__ __launch_bounds__(512) void reduce512_kernel(
    const float* __restrict__ partial, float* __restrict__ out)
{
    __shared__ float s[512];
    const int tid = threadIdx.x;
    s[tid] = partial[tid];
    __syncthreads();
    #pragma unroll
    for (int off = 256; off >= 1; off >>= 1) {
        if (tid < off) s[tid] += s[tid + off];
        __syncthreads();
    }
    if (tid == 0) out[0] = s[0] * (1.0f / (float)(T_DIM - 1));
}

extern "C" void kernel_launch(void* const* d_in, const int* in_sizes, int n_in,
                              void* d_out, int out_size, void* d_ws, size_t ws_size,
                              hipStream_t stream) {
    (void)in_sizes; (void)n_in; (void)out_size; (void)ws_size;
    const float* x = (const float*)d_in[0];   // [8192, 64]
    const float* w = (const float*)d_in[1];   // [8192]
    float* partial = (float*)d_ws;            // 512 floats
    float* out     = (float*)d_out;           // scalar

    const int rowTiles = T_DIM / 16;          // 512 waves
    band_wmma_kernel<<<rowTiles / WPB, WPB * 32, 0, stream>>>(x, w, partial);
    reduce512_kernel<<<1, 512, 0, stream>>>(partial, out);
}